// DeepResidualGCN_90958817394878
// MI455X (gfx1250) — compile-verified
//
#include <hip/hip_runtime.h>
#include <hip/hip_bf16.h>

// ---------------- problem constants (match reference) ----------------
#define NN 100000
#define NE 1600000
#define HH 64
#define CC 40
#define EPSBN 1e-5f

typedef __attribute__((ext_vector_type(2))) float v2f;
typedef __attribute__((ext_vector_type(8))) float v8f;

// ---------------- small helpers ----------------
__device__ __forceinline__ void gatomic_add(float* p, float v) {
    __hip_atomic_fetch_add(p, v, __ATOMIC_RELAXED, __HIP_MEMORY_SCOPE_AGENT);
}
__device__ __forceinline__ void latomic_add(float* p, float v) {
    __hip_atomic_fetch_add(p, v, __ATOMIC_RELAXED, __HIP_MEMORY_SCOPE_WORKGROUP);
}

// ---------------- fill ----------------
__global__ __launch_bounds__(256) void fill_f32(float* __restrict__ p, float v, long long n) {
    long long i = (long long)blockIdx.x * 256 + threadIdx.x;
    long long stride = (long long)gridDim.x * 256;
    for (; i < n; i += stride) p[i] = v;
}
static inline void launch_fill(float* p, float v, long long n, hipStream_t s) {
    long long b = (n + 255) / 256;
    int blocks = (b > 8192) ? 8192 : (int)(b < 1 ? 1 : b);
    fill_f32<<<blocks, 256, 0, s>>>(p, v, n);
}

// ---------------- graph setup ----------------
__global__ __launch_bounds__(256) void deg_kernel(const int* __restrict__ src,
                                                  float* __restrict__ deg, int nE) {
    int e = blockIdx.x * 256 + threadIdx.x;
    if (e < nE) gatomic_add(&deg[src[e]], 1.0f);
}
__global__ __launch_bounds__(256) void dis_kernel(const float* __restrict__ deg,
                                                  float* __restrict__ dis, int n) {
    int i = blockIdx.x * 256 + threadIdx.x;
    if (i < n) {
        float d = deg[i];
        dis[i] = (d > 0.0f) ? rsqrtf(d) : 0.0f;
    }
}
__global__ __launch_bounds__(256) void norm_kernel(const int* __restrict__ src,
                                                   const int* __restrict__ dst,
                                                   const float* __restrict__ dis,
                                                   float* __restrict__ normw, int nE) {
    int e = blockIdx.x * 256 + threadIdx.x;
    if (e < nE) normw[e] = -dis[src[e]] * dis[dst[e]];
}

// ---------------- sparse propagation: out[dst] += norm * t[src] (64 feats) ----------------
// thread = (edge, 16B chunk); 16 chunks of float4 cover the 64 features.
__global__ __launch_bounds__(256) void scatter_prop(const float* __restrict__ t,
                                                    const float* __restrict__ normw,
                                                    const int* __restrict__ src,
                                                    const int* __restrict__ dst,
                                                    float* __restrict__ out, int nE) {
    long long tid = (long long)blockIdx.x * 256 + threadIdx.x;
    int e = (int)(tid >> 4);
    int q = (int)(tid & 15);
    if (e >= nE) return;
    float w = normw[e];
    int s = src[e], d = dst[e];
    const float4 v = *(const float4*)(t + (size_t)s * HH + q * 4);
    float* o = out + (size_t)d * HH + q * 4;
    gatomic_add(o + 0, w * v.x);
    gatomic_add(o + 1, w * v.y);
    gatomic_add(o + 2, w * v.z);
    gatomic_add(o + 3, w * v.w);
}

// ---------------- WMMA f32 GEMM: Out (+)= (s1*A1 + s2*A2) @ W [+ bias] ----------------
// A: [nrows x 64] row-major. W: [64 x ncols] row-major. NT = #16-wide N tiles.
// DUAL/ACC/BIAS are compile-time so the inner loop is branch-free:
//   plain path per k-step: 1x global_load_b64 (A) + 2x ds_load_2addr_b64 (B) + NT x wmma.
// Per ISA 7.12.2 (32-bit): A(m,k): lane=m+16*(k>>1), vgpr=k&1. B(k,n): lane=n+16*(k>>1), vgpr=k&1.
// C/D: vgpr j: lanes 0-15 -> M=j, lanes 16-31 -> M=8+j.
template <int NT, bool DUAL, bool ACC, bool BIAS>
__global__ __launch_bounds__(256)
void gemm64_wmma(const float* __restrict__ A1, const float* __restrict__ A2,
                 float s1, float s2,
                 const float* __restrict__ W, int ncols,
                 const float* __restrict__ bias,
                 float* __restrict__ Out, int nrows) {
    // sW[kp*64+n] = { W[2kp][n], W[2kp+1][n] }  (zero-padded for n >= ncols)
    __shared__ v2f sW[32 * 64];
    const int tid = threadIdx.x;
    for (int i = tid; i < 32 * 64; i += 256) {
        int kp = i >> 6, n = i & 63;
        v2f w;
        w.x = (n < ncols) ? W[(2 * kp) * ncols + n] : 0.0f;
        w.y = (n < ncols) ? W[(2 * kp + 1) * ncols + n] : 0.0f;
        sW[i] = w;
    }
    __syncthreads();

    const int wave = tid >> 5;
    const int lane = tid & 31;
    const int rowBase = blockIdx.x * 128 + wave * 16;
    const int m = lane & 15;
    const int khalf = lane >> 4;  // 0 or 1
    int row = rowBase + m;
    int rowc = (row < nrows) ? row : (nrows - 1);  // clamp: loads in-bounds, EXEC all-ones

    v8f c[NT] = {};

    const float* a1p = A1 + (size_t)rowc * HH;
    const float* a2p = DUAL ? (A2 + (size_t)rowc * HH) : nullptr;

#pragma unroll
    for (int k0 = 0; k0 < 64; k0 += 4) {
        const int kb = k0 + khalf * 2;  // this lane's K pair
        float2 a1 = *(const float2*)(a1p + kb);
        v2f a;
        if (DUAL) {
            float2 a2 = *(const float2*)(a2p + kb);
            a.x = s1 * a1.x + s2 * a2.x;
            a.y = s1 * a1.y + s2 * a2.y;
        } else {
            a.x = a1.x;
            a.y = a1.y;
        }
        const int kp = kb >> 1;
#pragma unroll
        for (int t = 0; t < NT; ++t) {
            const int n = (lane & 15) + 16 * t;
            v2f b = sW[kp * 64 + n];
            c[t] = __builtin_amdgcn_wmma_f32_16x16x4_f32(
                false, a, false, b, (short)0, c[t], false, false);
        }
    }

#pragma unroll
    for (int t = 0; t < NT; ++t) {
        const int n = (lane & 15) + 16 * t;
        float bv = 0.0f;
        if (BIAS) bv = (n < ncols) ? bias[n] : 0.0f;
#pragma unroll
        for (int j = 0; j < 8; ++j) {
            const int r = rowBase + j + 8 * khalf;
            if (r < nrows && n < ncols) {
                const size_t idx = (size_t)r * ncols + n;
                float v = c[t][j];
                if (ACC) v += Out[idx];
                if (BIAS) v += bv;
                Out[idx] = v;
            }
        }
    }
}

// ---------------- BatchNorm stats: stats[c]=sum, stats[64+c]=sumsq ----------------
__global__ __launch_bounds__(256) void bn_stats(const float* __restrict__ h,
                                                float* __restrict__ stats,
                                                int nrows, int ncols) {
    __shared__ float ssum[64], ssq[64];
    const int tid = threadIdx.x;
    for (int cidx = tid; cidx < 64; cidx += 256) { ssum[cidx] = 0.0f; ssq[cidx] = 0.0f; }
    __syncthreads();

    long long row0 = (long long)blockIdx.x * 256;
    long long rows = (long long)nrows - row0;
    if (rows > 256) rows = 256;
    if (rows > 0) {
        const long long elems = rows * ncols;
        const float* base = h + row0 * ncols;
        float ls = 0.0f, lq = 0.0f;
        int cur = -1;
        for (long long i = tid; i < elems; i += 256) {
            int col = (int)(i % ncols);
            float v = base[i];
            if (col != cur) {
                if (cur >= 0) { latomic_add(&ssum[cur], ls); latomic_add(&ssq[cur], lq); }
                cur = col; ls = 0.0f; lq = 0.0f;
            }
            ls += v;
            lq += v * v;
        }
        if (cur >= 0) { latomic_add(&ssum[cur], ls); latomic_add(&ssq[cur], lq); }
    }
    __syncthreads();
    for (int cidx = tid; cidx < ncols; cidx += 256) {
        gatomic_add(&stats[cidx], ssum[cidx]);
        gatomic_add(&stats[64 + cidx], ssq[cidx]);
    }
}

// ---------------- fused BN(affine)+ReLU (+residual) ----------------
__global__ __launch_bounds__(256) void bn_relu_res(const float* __restrict__ h,
                                                   const float* __restrict__ stats,
                                                   const float* __restrict__ gamma,
                                                   const float* __restrict__ beta,
                                                   const float* __restrict__ ident,
                                                   float* __restrict__ out,
                                                   long long total, int ncols, float invN) {
    long long i = (long long)blockIdx.x * 256 + threadIdx.x;
    if (i >= total) return;
    int cidx = (int)(i % ncols);
    float mean = stats[cidx] * invN;
    float var = fmaf(-mean, mean, stats[64 + cidx] * invN);
    float v = (h[i] - mean) * rsqrtf(var + EPSBN) * gamma[cidx] + beta[cidx];
    v = fmaxf(v, 0.0f);
    if (ident) v += ident[i];
    out[i] = v;
}

// ---------------- log_softmax over 40 cols, one wave32 per row ----------------
__global__ __launch_bounds__(256) void log_softmax40(const float* __restrict__ h,
                                                     float* __restrict__ out, int nrows) {
    const int wave = threadIdx.x >> 5;
    const int lane = threadIdx.x & 31;
    const int row = blockIdx.x * 8 + wave;
    if (row >= nrows) return;
    const float* p = h + (size_t)row * CC;
    float v0 = p[lane];
    float v1 = (lane < 8) ? p[32 + lane] : -1e30f;
    float mx = fmaxf(v0, v1);
    for (int off = 16; off > 0; off >>= 1) mx = fmaxf(mx, __shfl_xor(mx, off, 32));
    float sm = expf(v0 - mx) + ((lane < 8) ? expf(v1 - mx) : 0.0f);
    for (int off = 16; off > 0; off >>= 1) sm += __shfl_xor(sm, off, 32);
    float L = mx + logf(sm);
    float* o = out + (size_t)row * CC;
    o[lane] = v0 - L;
    if (lane < 8) o[32 + lane] = v1 - L;
}

// ---------------- host orchestration ----------------
extern "C" void kernel_launch(void* const* d_in, const int* in_sizes, int n_in,
                              void* d_out, int out_size, void* d_ws, size_t ws_size,
                              hipStream_t stream) {
    (void)in_sizes; (void)n_in; (void)out_size; (void)ws_size;

    const float* x0   = (const float*)d_in[0];
    const int*   ei   = (const int*)d_in[1];
    const float* Wc   = (const float*)d_in[2];   // [3,3,64,64]
    // d_in[3] = bc_stack: cancelled exactly by BN mean subtraction -> unused
    const float* gam  = (const float*)d_in[4];   // [3,64]
    const float* bet  = (const float*)d_in[5];   // [3,64]
    const float* Wm   = (const float*)d_in[6];   // [3,64,64]
    const float* bm   = (const float*)d_in[7];   // [3,64]
    const float* WcL  = (const float*)d_in[8];   // [3,64,40]
    // d_in[9] = bc_last: cancelled by BN -> unused
    const float* gamL = (const float*)d_in[10];  // [40]
    const float* betL = (const float*)d_in[11];  // [40]
    float* out = (float*)d_out;

    const int* src = ei;
    const int* dst = ei + NE;

    // workspace layout (floats)
    float* ws    = (float*)d_ws;
    float* xbuf  = ws;                       // N*64
    float* tx1   = xbuf  + (size_t)NN * HH;  // N*64
    float* p2    = tx1   + (size_t)NN * HH;  // N*64
    float* ident = p2    + (size_t)NN * HH;  // N*64
    float* hout  = ident + (size_t)NN * HH;  // N*64
    float* deg   = hout  + (size_t)NN * HH;  // N
    float* dis   = deg   + NN;               // N
    float* normw = dis   + NN;               // E
    float* stats = normw + NE;               // 128
    float* hlast = stats + 128;              // N*40

    const int GB   = (NN + 127) / 128;          // gemm blocks (128 rows/block)
    const int EB   = (NE + 255) / 256;          // edge-parallel blocks
    const int NB   = (NN + 255) / 256;          // node-parallel blocks
    const int SB   = (int)(((long long)NE * 16 + 255) / 256);  // scatter blocks
    const float invN = 1.0f / (float)NN;

    // graph normalization setup
    launch_fill(deg, 0.0f, NN, stream);
    deg_kernel<<<EB, 256, 0, stream>>>(src, deg, NE);
    dis_kernel<<<NB, 256, 0, stream>>>(deg, dis, NN);
    norm_kernel<<<EB, 256, 0, stream>>>(src, dst, dis, normw, NE);

    const float* x = x0;
    for (int i = 0; i < 3; ++i) {
        const float* Wci = Wc + (size_t)i * 3 * HH * HH;
        // residual: ident = x @ Wm[i] + bm[i]
        gemm64_wmma<4, false, false, true><<<GB, 256, 0, stream>>>(
            x, nullptr, 1.0f, 0.0f, Wm + (size_t)i * HH * HH, HH,
            bm + (size_t)i * HH, ident, NN);
        // cheb k=0: hout = x @ Wc[i,0]
        gemm64_wmma<4, false, false, false><<<GB, 256, 0, stream>>>(
            x, nullptr, 1.0f, 0.0f, Wci, HH, nullptr, hout, NN);
        // T1 = prop(x); hout += T1 @ Wc[i,1]
        launch_fill(tx1, 0.0f, (long long)NN * HH, stream);
        scatter_prop<<<SB, 256, 0, stream>>>(x, normw, src, dst, tx1, NE);
        gemm64_wmma<4, false, true, false><<<GB, 256, 0, stream>>>(
            tx1, nullptr, 1.0f, 0.0f, Wci + (size_t)HH * HH, HH, nullptr, hout, NN);
        // P = prop(T1); hout += (2P - x) @ Wc[i,2]
        launch_fill(p2, 0.0f, (long long)NN * HH, stream);
        scatter_prop<<<SB, 256, 0, stream>>>(tx1, normw, src, dst, p2, NE);
        gemm64_wmma<4, true, true, false><<<GB, 256, 0, stream>>>(
            p2, x, 2.0f, -1.0f, Wci + (size_t)2 * HH * HH, HH, nullptr, hout, NN);
        // BN -> relu -> + ident
        launch_fill(stats, 0.0f, 128, stream);
        bn_stats<<<NB, 256, 0, stream>>>(hout, stats, NN, HH);
        bn_relu_res<<<(int)(((long long)NN * HH + 255) / 256), 256, 0, stream>>>(
            hout, stats, gam + (size_t)i * HH, bet + (size_t)i * HH, ident, xbuf,
            (long long)NN * HH, HH, invN);
        x = xbuf;
    }

    // last layer (out=40, no residual)
    gemm64_wmma<3, false, false, false><<<GB, 256, 0, stream>>>(
        x, nullptr, 1.0f, 0.0f, WcL, CC, nullptr, hlast, NN);
    launch_fill(tx1, 0.0f, (long long)NN * HH, stream);
    scatter_prop<<<SB, 256, 0, stream>>>(x, normw, src, dst, tx1, NE);
    gemm64_wmma<3, false, true, false><<<GB, 256, 0, stream>>>(
        tx1, nullptr, 1.0f, 0.0f, WcL + (size_t)HH * CC, CC, nullptr, hlast, NN);
    launch_fill(p2, 0.0f, (long long)NN * HH, stream);
    scatter_prop<<<SB, 256, 0, stream>>>(tx1, normw, src, dst, p2, NE);
    gemm64_wmma<3, true, true, false><<<GB, 256, 0, stream>>>(
        p2, x, 2.0f, -1.0f, WcL + (size_t)2 * HH * CC, CC, nullptr, hlast, NN);
    launch_fill(stats, 0.0f, 128, stream);
    bn_stats<<<NB, 256, 0, stream>>>(hlast, stats, NN, CC);
    bn_relu_res<<<(int)(((long long)NN * CC + 255) / 256), 256, 0, stream>>>(
        hlast, stats, gamL, betL, nullptr, hlast, (long long)NN * CC, CC, invN);
    log_softmax40<<<(NN + 7) / 8, 256, 0, stream>>>(hlast, out, NN);
}